// MolEncoderFT_84078279786625
// MI455X (gfx1250) — compile-verified
//
#include <hip/hip_runtime.h>

// ---------------------------------------------------------------------------
// MolEncoderFT for MI455X (gfx1250, wave32, WMMA)
//
// reference pipeline:
//   hn = h_node @ W_node ; he = h_edge @ W_edge      (linear)
//   mean per graph (contiguous equal segments: 50 nodes / 100 edges)
//   emb  = relu([mean_n, mean_e] @ W1 + b1) @ W2 + b2
//   pred = emb[:, :64] @ Wa + ba
//
// optimization: linearity lets us commute mean with the embedders AND fold
// the embedders into W1:  Wn1 = W_node @ W1[:128], We1 = W_edge @ W1[128:].
// Per-graph MLP input shrinks from 256 features to 16+5 raw means.
// HBM-bound part is the 52 MB segment reduction (~2.3 us at 23.3 TB/s),
// streamed with b128 loads; the two MLP layers run on V_WMMA_F32_16X16X4_F32.
// ---------------------------------------------------------------------------

typedef float v2f __attribute__((ext_vector_type(2)));
typedef float v8f __attribute__((ext_vector_type(8)));

static constexpr int NG          = 10000;
static constexpr int N_NODES     = 500000;
static constexpr int NODE_FLOATS = 16;        // h_node cols
static constexpr int EMB_DIM     = 128;
static constexpr int H1_PITCH    = 260;       // stride % 64 == 4 -> conflict-free
static constexpr int EMB_PITCH   = 132;

// ---------------------------------------------------------------------------
// Kernel 1: per-graph means of raw h_node (16 cols) and h_edge (5 cols).
// One wave per graph.
// Nodes: 800 floats = 200 aligned float4 chunks; lane walks chunks with
// stride 32, so its 4-column group is fixed at 4*(lane&3). 7 b128 loads per
// lane (512 B/wave/instruction -> max bytes in flight per LOADcnt slot),
// then a 3-step shfl_xor tree over the 8 lanes sharing a group, and a single
// float4 store from lanes 0..3.
// Edges: 100 rows x 5 floats (rows not 16B-aligned) -> lane-per-row scalar
// loads (wave still reads 640 B contiguous), 5-component shfl_xor tree.
// ---------------------------------------------------------------------------
__global__ void k_segmean(const float* __restrict__ h_node,
                          const float* __restrict__ h_edge,
                          float* __restrict__ sn,      // [NG][16]
                          float* __restrict__ se)      // [NG][8] (cols 5..7 = 0)
{
    int wave = (int)((blockIdx.x * blockDim.x + threadIdx.x) >> 5);
    int lane = (int)(threadIdx.x & 31);
    if (wave >= NG) return;

    // ---- nodes: 50 rows x 16 cols = 800 floats = 200 float4 ----
    const float4* np4 = (const float4*)(h_node + (size_t)wave * 800);
    float ax = 0.f, ay = 0.f, az = 0.f, aw = 0.f;
#pragma unroll
    for (int t = 0; t < 7; ++t) {
        int j = lane + 32 * t;               // chunk index; j%4 == lane%4
        if (j < 200) {
            float4 v = np4[j];
            ax += v.x; ay += v.y; az += v.z; aw += v.w;
        }
    }
#pragma unroll
    for (int off = 16; off >= 4; off >>= 1) {   // reduce lanes sharing lane&3
        ax += __shfl_xor(ax, off, 32);
        ay += __shfl_xor(ay, off, 32);
        az += __shfl_xor(az, off, 32);
        aw += __shfl_xor(aw, off, 32);
    }
    if (lane < 4) {                          // lane g owns cols 4g..4g+3
        float4 o;
        o.x = ax * (1.0f / 50.0f);
        o.y = ay * (1.0f / 50.0f);
        o.z = az * (1.0f / 50.0f);
        o.w = aw * (1.0f / 50.0f);
        *(float4*)(sn + (size_t)wave * NODE_FLOATS + 4 * lane) = o;
    }

    // ---- edges: 100 rows x 5 cols = 500 floats ----
    const float* ep = h_edge + (size_t)wave * 500;
    float e0 = 0.f, e1 = 0.f, e2 = 0.f, e3 = 0.f, e4 = 0.f;
    for (int r = lane; r < 100; r += 32) {
        const float* row = ep + r * 5;
        e0 += row[0]; e1 += row[1]; e2 += row[2]; e3 += row[3]; e4 += row[4];
    }
#pragma unroll
    for (int off = 16; off >= 1; off >>= 1) {
        e0 += __shfl_xor(e0, off, 32);
        e1 += __shfl_xor(e1, off, 32);
        e2 += __shfl_xor(e2, off, 32);
        e3 += __shfl_xor(e3, off, 32);
        e4 += __shfl_xor(e4, off, 32);
    }
    if (lane == 0) {
        float* o = se + (size_t)wave * 8;
        o[0] = e0 * 0.01f; o[1] = e1 * 0.01f; o[2] = e2 * 0.01f;
        o[3] = e3 * 0.01f; o[4] = e4 * 0.01f;
        o[5] = 0.0f; o[6] = 0.0f; o[7] = 0.0f;   // zero-pad K to 8
    }
}

// ---------------------------------------------------------------------------
// Kernel 2: fold embedders into MLP layer 1.
//   Wn1[16][256] = W_node[16][128] @ W1[0:128]
//   We1[8][256]  = W_edge[5][128]  @ W1[128:256]  (rows 5..7 zeroed)
// One block of 256 threads, thread n owns output column n.
// ---------------------------------------------------------------------------
__global__ void k_fold(const float* __restrict__ Wnode,
                       const float* __restrict__ Wedge,
                       const float* __restrict__ W1,
                       float* __restrict__ Wn1,
                       float* __restrict__ We1)
{
    int n = (int)threadIdx.x;               // 0..255
    float accN[16];
#pragma unroll
    for (int r = 0; r < 16; ++r) accN[r] = 0.0f;
    for (int k = 0; k < 128; ++k) {
        float w = W1[(size_t)k * 256 + n];  // coalesced
#pragma unroll
        for (int r = 0; r < 16; ++r)        // W_node reads are wave-uniform
            accN[r] += Wnode[r * 128 + k] * w;
    }
#pragma unroll
    for (int r = 0; r < 16; ++r) Wn1[r * 256 + n] = accN[r];

    float accE[5];
#pragma unroll
    for (int r = 0; r < 5; ++r) accE[r] = 0.0f;
    for (int k = 0; k < 128; ++k) {
        float w = W1[(size_t)(128 + k) * 256 + n];
#pragma unroll
        for (int r = 0; r < 5; ++r)
            accE[r] += Wedge[r * 128 + k] * w;
    }
#pragma unroll
    for (int r = 0; r < 5; ++r) We1[r * 256 + n] = accE[r];
#pragma unroll
    for (int r = 5; r < 8; ++r) We1[r * 256 + n] = 0.0f;
}

// ---------------------------------------------------------------------------
// Kernel 3: batch_node passthrough (int32 -> float), vectorized int4/float4.
// N_NODES == 500000 is divisible by 4.
// ---------------------------------------------------------------------------
__global__ void k_batch(const int* __restrict__ b, float* __restrict__ out, int n4)
{
    int i = (int)(blockIdx.x * blockDim.x + threadIdx.x);
    if (i < n4) {
        int4 v = ((const int4*)b)[i];
        float4 f;
        f.x = (float)v.x; f.y = (float)v.y; f.z = (float)v.z; f.w = (float)v.w;
        ((float4*)out)[i] = f;
    }
}

// ---------------------------------------------------------------------------
// Kernel 4: fused MLP + head, one wave (32 threads) per 16-graph M-tile,
// layers 1-2 via V_WMMA_F32_16X16X4_F32, head via plain FMA.
//
// A 16x4 f32 layout (ISA 7.12.2): lane l covers row m=l&15;
//   v[0]=A[m][k0 + (l<16?0:2)], v[1]=A[m][k0 + (l<16?1:3)]
// B 4x16 f32: lane l covers col n=l&15;
//   v[0]=B[k0 + (l<16?0:2)][n], v[1]=B[k0 + (l<16?1:3)][n]
// C/D 16x16: lane l col n=l&15, vgpr i row m = i + (l<16?0:8).
// ---------------------------------------------------------------------------
__global__ void __launch_bounds__(32) k_mlp(
    const float* __restrict__ sn,   // [NG][16]
    const float* __restrict__ se,   // [NG][8]
    const float* __restrict__ Wn1,  // [16][256]
    const float* __restrict__ We1,  // [8][256]
    const float* __restrict__ b1,   // [256]
    const float* __restrict__ W2,   // [256][128]
    const float* __restrict__ b2,   // [128]
    const float* __restrict__ Wa,   // [64][4]
    const float* __restrict__ ba,   // [4]
    float* __restrict__ emb_out,    // [NG][128]
    float* __restrict__ pred_out)   // [NG][4]
{
    __shared__ float h1[16 * H1_PITCH];     // layer-1 activations (A of layer 2)
    __shared__ float embs[16 * EMB_PITCH];  // emb (A of the head)

    const int m0   = (int)blockIdx.x * 16;  // first graph of this tile
    const int l    = (int)threadIdx.x;      // 0..31, full wave (EXEC all 1s)
    const int col  = l & 15;
    const int hi   = (l >= 16) ? 1 : 0;
    const int krow = hi * 2;                // K offset for this half-wave

    // -------- layer 1: [16 x 24] @ [24 x 256] (+b1, relu) --------
    for (int nt = 0; nt < 16; ++nt) {
        const int n0 = nt * 16;
        v8f c = {0.f, 0.f, 0.f, 0.f, 0.f, 0.f, 0.f, 0.f};
#pragma unroll
        for (int ks = 0; ks < 4; ++ks) {            // K = 16 over node means
            const int k0 = ks * 4;
            v2f a, b;
            a.x = sn[(size_t)(m0 + col) * 16 + k0 + krow];
            a.y = sn[(size_t)(m0 + col) * 16 + k0 + krow + 1];
            b.x = Wn1[(size_t)(k0 + krow)     * 256 + n0 + col];
            b.y = Wn1[(size_t)(k0 + krow + 1) * 256 + n0 + col];
            c = __builtin_amdgcn_wmma_f32_16x16x4_f32(false, a, false, b,
                                                      (short)0, c, false, false);
        }
#pragma unroll
        for (int ks = 0; ks < 2; ++ks) {            // K = 8 over edge means (padded)
            const int k0 = ks * 4;
            v2f a, b;
            a.x = se[(size_t)(m0 + col) * 8 + k0 + krow];
            a.y = se[(size_t)(m0 + col) * 8 + k0 + krow + 1];
            b.x = We1[(size_t)(k0 + krow)     * 256 + n0 + col];
            b.y = We1[(size_t)(k0 + krow + 1) * 256 + n0 + col];
            c = __builtin_amdgcn_wmma_f32_16x16x4_f32(false, a, false, b,
                                                      (short)0, c, false, false);
        }
        const float bias = b1[n0 + col];
#pragma unroll
        for (int i = 0; i < 8; ++i) {
            float v = c[i] + bias;
            v = v > 0.0f ? v : 0.0f;                // relu
            h1[(i + hi * 8) * H1_PITCH + n0 + col] = v;
        }
    }
    __syncthreads();   // single wave; DS in-order, but keep the fence explicit

    // -------- layer 2: [16 x 256] @ [256 x 128] (+b2) --------
    for (int nt = 0; nt < 8; ++nt) {
        const int n0 = nt * 16;
        v8f c = {0.f, 0.f, 0.f, 0.f, 0.f, 0.f, 0.f, 0.f};
        for (int ks = 0; ks < 64; ++ks) {
            const int k0 = ks * 4;
            // A from LDS: float2 (8B aligned: pitch even, k0+krow even)
            v2f a = *(const v2f*)&h1[col * H1_PITCH + k0 + krow];
            v2f b;
            b.x = W2[(size_t)(k0 + krow)     * 128 + n0 + col];
            b.y = W2[(size_t)(k0 + krow + 1) * 128 + n0 + col];
            c = __builtin_amdgcn_wmma_f32_16x16x4_f32(false, a, false, b,
                                                      (short)0, c, false, false);
        }
        const float bias = b2[n0 + col];
#pragma unroll
        for (int i = 0; i < 8; ++i) {
            const float v = c[i] + bias;
            const int   m = i + hi * 8;
            emb_out[(size_t)(m0 + m) * EMB_DIM + n0 + col] = v;
            embs[m * EMB_PITCH + n0 + col] = v;
        }
    }
    __syncthreads();

    // -------- head: pred = emb[:, :64] @ Wa + ba  (tiny: plain FMA) --------
    if (l < 16) {
        float p0 = ba[0], p1 = ba[1], p2 = ba[2], p3 = ba[3];
        for (int k = 0; k < 64; ++k) {
            const float x = embs[l * EMB_PITCH + k];
            p0 += x * Wa[k * 4 + 0];
            p1 += x * Wa[k * 4 + 1];
            p2 += x * Wa[k * 4 + 2];
            p3 += x * Wa[k * 4 + 3];
        }
        float* o = pred_out + (size_t)(m0 + l) * 4;
        o[0] = p0; o[1] = p1; o[2] = p2; o[3] = p3;
    }
}

// ---------------------------------------------------------------------------
// Host launcher
// ---------------------------------------------------------------------------
extern "C" void kernel_launch(void* const* d_in, const int* in_sizes, int n_in,
                              void* d_out, int out_size, void* d_ws, size_t ws_size,
                              hipStream_t stream)
{
    (void)in_sizes; (void)n_in; (void)out_size; (void)ws_size;

    const float* h_node = (const float*)d_in[0];
    // d_in[1] = pos_node (unused by reference math)
    const float* h_edge = (const float*)d_in[2];
    const float* W_node = (const float*)d_in[3];
    const float* W_edge = (const float*)d_in[4];
    const float* W1     = (const float*)d_in[5];
    const float* b1     = (const float*)d_in[6];
    const float* W2     = (const float*)d_in[7];
    const float* b2     = (const float*)d_in[8];
    const float* Wa     = (const float*)d_in[9];
    const float* ba     = (const float*)d_in[10];
    const int*   batch_node = (const int*)d_in[11];
    // d_in[12] = edge_index, d_in[13] = batch_edge (unused by reference math)

    float* out = (float*)d_out;
    float* emb_out   = out;                                 // [10000][128]
    float* batch_out = out + (size_t)NG * EMB_DIM;          // [500000]
    float* pred_out  = batch_out + N_NODES;                 // [10000][4]

    // workspace layout (floats): sn[10000*16] | se[10000*8] | Wn1[16*256] | We1[8*256]
    float* ws  = (float*)d_ws;
    float* sn  = ws;
    float* se  = ws + (size_t)NG * NODE_FLOATS;
    float* Wn1 = se + (size_t)NG * 8;
    float* We1 = Wn1 + 16 * 256;

    // 1) segment means: one wave per graph, 8 waves per block
    k_segmean<<<dim3((NG * 32 + 255) / 256), dim3(256), 0, stream>>>(
        h_node, h_edge, sn, se);

    // 2) fold embedders into layer-1 weights
    k_fold<<<dim3(1), dim3(256), 0, stream>>>(W_node, W_edge, W1, Wn1, We1);

    // 3) batch_node passthrough to output (float4 granularity)
    k_batch<<<dim3((N_NODES / 4 + 255) / 256), dim3(256), 0, stream>>>(
        batch_node, batch_out, N_NODES / 4);

    // 4) fused MLP + head, 16 graphs per wave, WMMA f32
    k_mlp<<<dim3(NG / 16), dim3(32), 0, stream>>>(
        sn, se, Wn1, We1, b1, W2, b2, Wa, ba, emb_out, pred_out);
}